// RoIPool_23845658427939
// MI455X (gfx1250) — compile-verified
//
#include <hip/hip_runtime.h>
#include <hip/hip_bf16.h>
#include <math.h>

// Problem constants (from the reference setup)
constexpr int   kB = 4;
constexpr int   kC = 256;
constexpr int   kH = 50;
constexpr int   kW = 50;
constexpr int   kN = 512;
constexpr int   kP = 7;                       // OUTPUT_SIZE
constexpr float kScale = 0.0625f;             // 1/16
constexpr int   kPerRoi = kC * kP * kP;       // 12544
constexpr int   kBlock  = 256;                // 8 wave32 waves
constexpr int   kBlocksPerRoi = kPerRoi / kBlock;   // 49 exactly
constexpr int   kGrid = kN * kBlocksPerRoi;   // 25088 blocks
constexpr int   kTotal = kN * kPerRoi;        // 6,422,528 outputs

// gfx1250 async global->LDS path (ASYNCcnt), guarded so compile never breaks.
#if defined(__gfx1250__) && \
    __has_builtin(__builtin_amdgcn_global_load_async_to_lds_b32) && \
    __has_builtin(__builtin_amdgcn_s_wait_asynccnt)
#define ROIPOOL_ASYNC_LDS 1
typedef __attribute__((address_space(1))) int gas_int;   // global (AS1) int
typedef __attribute__((address_space(3))) int lds_int;   // LDS (AS3) int
#else
#define ROIPOOL_ASYNC_LDS 0
#endif

__global__ __launch_bounds__(kBlock) void roi_pool_cdna5_kernel(
    const float* __restrict__ input,   // [B,C,H,W]
    const float* __restrict__ rois,    // [N,5] : (b, x1, y1, x2, y2) in image coords
    float* __restrict__ out)           // [N,C,PH,PW]
{
    __shared__ float sroi[8];

    const int lane = threadIdx.x;
    const int n    = blockIdx.x / kBlocksPerRoi;   // uniform per block (SALU)

    // Stage the 5 ROI descriptor floats into LDS once per block.
#if ROIPOOL_ASYNC_LDS
    if (lane < 5) {
        __builtin_amdgcn_global_load_async_to_lds_b32(
            (gas_int*)(rois + n * 5 + lane),
            (lds_int*)(&sroi[lane]),
            /*imm offset*/ 0, /*cpol*/ 0);
    }
    __builtin_amdgcn_s_wait_asynccnt(0);           // wave-level wait (no-op for waves 1..7)
    __syncthreads();
#else
    if (lane < 5) sroi[lane] = rois[n * 5 + lane];
    __syncthreads();
#endif

    const int idx = blockIdx.x * kBlock + lane;    // grid sized exactly: idx < kTotal
    const int pw  = idx % kP;
    int t         = idx / kP;
    const int ph  = t % kP;
    t            /= kP;
    const int c   = t % kC;

    // mmcv RoIPool quantization (matches reference exactly)
    const int b  = (int)sroi[0];
    const int x1 = (int)floorf(sroi[1] * kScale + 0.5f);
    const int y1 = (int)floorf(sroi[2] * kScale + 0.5f);
    const int x2 = (int)floorf(sroi[3] * kScale + 0.5f);
    const int y2 = (int)floorf(sroi[4] * kScale + 0.5f);

    const int rw = max(x2 - x1 + 1, 1);
    const int rh = max(y2 - y1 + 1, 1);
    const float bin_h = (float)rh / (float)kP;
    const float bin_w = (float)rw / (float)kP;

    const int hstart = min(max((int)floorf(bin_h * (float)ph)       + y1, 0), kH);
    const int hend   = min(max((int)ceilf (bin_h * (float)(ph + 1)) + y1, 0), kH);
    const int wstart = min(max((int)floorf(bin_w * (float)pw)       + x1, 0), kW);
    const int wend   = min(max((int)ceilf (bin_w * (float)(pw + 1)) + x1, 0), kW);

    const float* __restrict__ base = input + (size_t)(b * kC + c) * (kH * kW);

    float maxv = -INFINITY;
    for (int h = hstart; h < hend; ++h) {
        const float* __restrict__ row = base + h * kW + wstart;
        // Prefetch next row of this bin while reducing the current one
        // (lowers to global_prefetch_b8 on gfx1250).
        if (h + 1 < hend) __builtin_prefetch(row + kW, 0, 0);
        const int span = wend - wstart;
        #pragma unroll 4
        for (int w = 0; w < span; ++w) {
            maxv = fmaxf(maxv, row[w]);
        }
    }

    // Empty bins -> 0 (reference replaces -inf with 0)
    out[idx] = (maxv == -INFINITY) ? 0.0f : maxv;
}

extern "C" void kernel_launch(void* const* d_in, const int* in_sizes, int n_in,
                              void* d_out, int out_size, void* d_ws, size_t ws_size,
                              hipStream_t stream) {
    (void)in_sizes; (void)n_in; (void)out_size; (void)d_ws; (void)ws_size;
    const float* input = (const float*)d_in[0];   // (4,256,50,50) f32
    const float* rois  = (const float*)d_in[1];   // (512,5)       f32
    float*       out   = (float*)d_out;           // (512,256,7,7) f32

    roi_pool_cdna5_kernel<<<dim3(kGrid), dim3(kBlock), 0, stream>>>(input, rois, out);
}